// Head_41798621724695
// MI455X (gfx1250) — compile-verified
//
#include <hip/hip_runtime.h>
#include <math.h>
#include <stdint.h>

// ---------------------------------------------------------------------------
// Head (single-head causal attention + RoPE) for MI455X / gfx1250.
//   B=4, T=4096, C=1024, H=64, fp32.
// GEMMs via V_WMMA_F32_16X16X4_F32 (fp32 in/accum -> reference numerics).
// Tile feeds via Tensor Data Mover (tensor_load_to_lds, TENSORcnt,
// double-buffered LDS with TDM pad for bank-conflict-free ds reads).
// ---------------------------------------------------------------------------

typedef float v2f __attribute__((ext_vector_type(2)));
typedef float v8f __attribute__((ext_vector_type(8)));
typedef unsigned int u32x4 __attribute__((ext_vector_type(4)));
typedef int i32x4 __attribute__((ext_vector_type(4)));
typedef int i32x8 __attribute__((ext_vector_type(8)));

#define B_  4
#define T_  4096
#define C_  1024
#define H_  64
#define TT  (T_ / 16)          // 256 row tiles per batch

__device__ __forceinline__ v8f wmma_f32(v2f a, v2f b, v8f c) {
  // D = A(16x4, f32) * B(4x16, f32) + C(16x16, f32)
  return __builtin_amdgcn_wmma_f32_16x16x4_f32(
      false, a, false, b, (short)0, c, false, false);
}

// TDM 2-D tile load: rows x row_elems fp32, row pitch row_stride elems in
// global memory; pad_ctl adds LDS padding after each row (ISA 08 D# group1).
// This toolchain's builtin takes 6 args (extra i32x8 descriptor words + cpol).
__device__ __forceinline__ void tdm_load_2d(
    unsigned lds_off, const float* gsrc,
    int row_elems, int rows, int row_stride, unsigned pad_ctl)
{
  unsigned long long ga = (unsigned long long)(uintptr_t)gsrc;
  u32x4 g0 = { 1u,                                   // count=1 (valid user D#)
               lds_off,                               // lds_addr
               (unsigned)ga,                          // global_addr[31:0]
               (unsigned)((ga >> 32) & 0x01FFFFFFu)   // global_addr[56:32]
                 | 0x80000000u };                     // type=2 ("image")
  i32x8 g1 = {
    (int)((2u << 16) | pad_ctl),                     // data_size=4B | pad bits
    (int)((unsigned)row_elems << 16),                // tensor_dim0[15:0]
    (int)(((unsigned)row_elems >> 16) | ((unsigned)rows << 16)), // dim0 hi|dim1 lo
    (int)(((unsigned)rows >> 16) | ((unsigned)row_elems << 16)), // dim1 hi|tile_dim0
    rows,                                            // tile_dim1 (tile_dim2=0)
    row_stride,                                      // tensor_dim0_stride[31:0]
    0, 0 };
  i32x4 z4 = { 0, 0, 0, 0 };                         // groups 2/3: 2-D tensor
  i32x8 z8 = { 0, 0, 0, 0, 0, 0, 0, 0 };
  __builtin_amdgcn_tensor_load_to_lds(g0, g1, z4, z4, z8, 0);
}

#define PAD_K1 ((1u << 20) | (5u << 22) | (3u << 25))  // interval 64 dw, pad 4 -> pitch 68
#define PAD_K2 ((1u << 20) | (6u << 22) | (3u << 25))  // interval 128 dw, pad 4 -> pitch 132

// ---------------------------------------------------------------------------
// Kernel 1: Q = rope(x@wq), KV[t] = [rope(x@wk) | x@wv] interleaved (pitch 128)
// One wave per (batch, 16-row tile); x streamed through LDS via TDM,
// double-buffered in 16x64 chunks; 12 WMMAs per K-step.
// ---------------------------------------------------------------------------
__global__ __launch_bounds__(128) void proj_rope_kernel(
    const float* __restrict__ x,  const float* __restrict__ wq,
    const float* __restrict__ wk, const float* __restrict__ wv,
    float* __restrict__ Q, float* __restrict__ KV)
{
  __shared__ __align__(16) float xbuf[4][2][16 * 68];  // per-wave double buffer

  const int lane = threadIdx.x & 31;
  const int wave = threadIdx.x >> 5;
  const int tile = blockIdx.x * 4 + wave;        // 0 .. B*TT-1
  const int b    = tile / TT;
  const int t0   = (tile % TT) * 16;
  const int lh   = lane >> 4;
  const int ll   = lane & 15;

  v8f qa[4], ka[4], va[4];
#pragma unroll
  for (int nt = 0; nt < 4; ++nt) { qa[nt] = (v8f)(0.f); ka[nt] = (v8f)(0.f); va[nt] = (v8f)(0.f); }

  const float* xbase = x + ((size_t)b * T_ + t0) * C_;

  tdm_load_2d((unsigned)(uintptr_t)&xbuf[wave][0][0], xbase, 64, 16, C_, PAD_K1);

  for (int c = 0; c < 16; ++c) {                 // 16 chunks of 64 K-values
    if (c < 15) {
      asm volatile("s_wait_dscnt 0" ::: "memory");
      tdm_load_2d((unsigned)(uintptr_t)&xbuf[wave][(c + 1) & 1][0],
                  xbase + (c + 1) * 64, 64, 16, C_, PAD_K1);
      __builtin_amdgcn_s_wait_tensorcnt((short)1);   // chunk c resident
    } else {
      __builtin_amdgcn_s_wait_tensorcnt((short)0);
    }
    asm volatile("" ::: "memory");

    const float* xl = &xbuf[wave][c & 1][0];
#pragma unroll 4
    for (int k4 = 0; k4 < 64; k4 += 4) {
      const v2f a = *(const v2f*)(xl + ll * 68 + k4 + 2 * lh);  // A: x[t0+ll, k+2lh+v]
      const int kk = (c * 64 + k4 + 2 * lh) * H_ + ll;
#pragma unroll
      for (int nt = 0; nt < 4; ++nt) {
        const int n0 = nt * 16;
        v2f bq = { wq[kk + n0], wq[kk + H_ + n0] };
        v2f bk = { wk[kk + n0], wk[kk + H_ + n0] };
        v2f bv = { wv[kk + n0], wv[kk + H_ + n0] };
        qa[nt] = wmma_f32(a, bq, qa[nt]);
        ka[nt] = wmma_f32(a, bk, ka[nt]);
        va[nt] = wmma_f32(a, bv, va[nt]);
      }
    }
  }

  // RoPE in-register: acc element r = (row t0+r+8*lh, col nt*16+ll);
  // pair partner (h^1) sits in lane^1 -> 1-lane xor shuffle.
  const float L2TH = 13.287712379549449f;        // log2(10000)
#pragma unroll
  for (int nt = 0; nt < 4; ++nt) {
    const int h = nt * 16 + ll;
    const float inv_freq = exp2f(-(float)(h & ~1) * (L2TH / (float)H_));
    const float sign = (h & 1) ? 1.f : -1.f;
#pragma unroll
    for (int r = 0; r < 8; ++r) {
      const int t = t0 + r + 8 * lh;
      const float ang = (float)t * inv_freq;
      const float cth = cosf(ang), sth = sinf(ang);

      float qv = qa[nt][r], qp = __shfl_xor(qv, 1, 32);
      qa[nt][r] = qv * cth + sign * qp * sth;    // even: a*c - b*s ; odd: b*c + a*s
      float kv = ka[nt][r], kp = __shfl_xor(kv, 1, 32);
      ka[nt][r] = kv * cth + sign * kp * sth;
    }
  }

  // Store Q [B,T,64] and interleaved KV [B,T,128] (K in 0..63, V in 64..127)
#pragma unroll
  for (int nt = 0; nt < 4; ++nt) {
    const int h = nt * 16 + ll;
#pragma unroll
    for (int r = 0; r < 8; ++r) {
      const int t = t0 + r + 8 * lh;
      Q[((size_t)b * T_ + t) * H_ + h]          = qa[nt][r];
      KV[((size_t)b * T_ + t) * 128 + h]        = ka[nt][r];
      KV[((size_t)b * T_ + t) * 128 + 64 + h]   = va[nt][r];
    }
  }
}

// ---------------------------------------------------------------------------
// Kernel 2: flash attention. One wave = one batch; each wave processes the
// balanced tile pair (pair, TT-1-pair) -> uniform ~TT work per wave. KV tiles
// (16 x 128 fp32) arrive via double-buffered TDM; no block barriers needed.
// ---------------------------------------------------------------------------
__global__ __launch_bounds__(64) void attn_kernel(
    const float* __restrict__ Q, const float* __restrict__ KV,
    float* __restrict__ O)
{
  __shared__ __align__(16) float kvb[2][2][16 * 132];  // per-wave double buffer
  __shared__ __align__(16) float pb[2][16][18];        // per-wave P transpose

  const int lane = threadIdx.x & 31;
  const int wave = threadIdx.x >> 5;             // 0..1
  const int b    = 2 * (blockIdx.x & 1) + wave;  // batch
  const int pair = blockIdx.x >> 1;              // 0..TT/2-1
  const int lh   = lane >> 4;
  const int ll   = lane & 15;

  const float* kvsrc = KV + (size_t)b * T_ * 128;

  for (int half = 0; half < 2; ++half) {
    const int tq = half ? (TT - 1 - pair) : pair;
    const int t0 = tq * 16;

    // Q A-fragments, resident for the tile: qf[s][v] = Q[t0+ll, 4s+2lh+v]
    v2f qf[16];
    {
      const float* qrow = Q + ((size_t)b * T_ + t0) * H_ + (size_t)ll * H_;
#pragma unroll
      for (int s = 0; s < 16; ++s) qf[s] = *(const v2f*)(qrow + 4 * s + 2 * lh);
    }

    v8f   oacc[4];
    float mrow[8], lrow[8];
#pragma unroll
    for (int nt = 0; nt < 4; ++nt) oacc[nt] = (v8f)(0.f);
#pragma unroll
    for (int r = 0; r < 8; ++r) { mrow[r] = -3.0e38f; lrow[r] = 0.f; }

    asm volatile("s_wait_dscnt 0" ::: "memory");
    tdm_load_2d((unsigned)(uintptr_t)&kvb[wave][0][0], kvsrc, 128, 16, 128, PAD_K2);

    for (int j = 0; j <= tq; ++j) {
      if (j < tq) {
        asm volatile("s_wait_dscnt 0" ::: "memory");
        tdm_load_2d((unsigned)(uintptr_t)&kvb[wave][(j + 1) & 1][0],
                    kvsrc + (size_t)(j + 1) * 16 * 128, 128, 16, 128, PAD_K2);
        __builtin_amdgcn_s_wait_tensorcnt((short)1);  // tile j resident
      } else {
        __builtin_amdgcn_s_wait_tensorcnt((short)0);
      }
      asm volatile("" ::: "memory");

      const float* kl = &kvb[wave][j & 1][0];    // row pitch 132 (TDM pad)

      // S = Q * K^T : B frag = K[tk+ll, 4st+2lh+v]
      v8f s = (v8f)(0.f);
#pragma unroll
      for (int st = 0; st < 16; ++st) {
        const v2f bf = *(const v2f*)(kl + ll * 132 + 4 * st + 2 * lh);
        s = wmma_f32(qf[st], bf, s);
      }

      // scale, causal mask, online softmax (row = r + 8*lh, col = ll)
#pragma unroll
      for (int r = 0; r < 8; ++r) {
        float sv = s[r] * 0.125f;                // 1/sqrt(H)
        if (j == tq && ll > (r + 8 * lh)) sv = -3.0e38f;

        float rmax = sv;
        rmax = fmaxf(rmax, __shfl_xor(rmax, 1, 32));
        rmax = fmaxf(rmax, __shfl_xor(rmax, 2, 32));
        rmax = fmaxf(rmax, __shfl_xor(rmax, 4, 32));
        rmax = fmaxf(rmax, __shfl_xor(rmax, 8, 32));

        const float mnew = fmaxf(mrow[r], rmax);
        const float sc   = __expf(mrow[r] - mnew);
        mrow[r] = mnew;

        const float p = __expf(sv - mnew);
        s[r] = p;

        float rs = p;
        rs += __shfl_xor(rs, 1, 32);
        rs += __shfl_xor(rs, 2, 32);
        rs += __shfl_xor(rs, 4, 32);
        rs += __shfl_xor(rs, 8, 32);
        lrow[r] = lrow[r] * sc + rs;

#pragma unroll
        for (int nt = 0; nt < 4; ++nt) oacc[nt][r] *= sc;
      }

      // Transpose P (C-layout -> A-layout) through wave-private LDS.
      // LDS is in-order per wave; fence + dscnt drain keep compiler honest.
#pragma unroll
      for (int r = 0; r < 8; ++r) pb[wave][r + 8 * lh][ll] = s[r];
      asm volatile("s_wait_dscnt 0" ::: "memory");

      // O += P * V   (V rows at kl[row*132 + 64 + h])
#pragma unroll
      for (int s2 = 0; s2 < 4; ++s2) {
        const int kk = 4 * s2 + 2 * lh;
        const v2f pa = *(const v2f*)&pb[wave][ll][kk];
#pragma unroll
        for (int nt = 0; nt < 4; ++nt) {
          const v2f vb = { kl[kk * 132 + 64 + nt * 16 + ll],
                           kl[(kk + 1) * 132 + 64 + nt * 16 + ll] };
          oacc[nt] = wmma_f32(pa, vb, oacc[nt]);
        }
      }
    }

    // Epilogue: normalize and store this tile
#pragma unroll
    for (int nt = 0; nt < 4; ++nt) {
#pragma unroll
      for (int r = 0; r < 8; ++r) {
        const int t = t0 + r + 8 * lh;
        O[((size_t)b * T_ + t) * H_ + nt * 16 + ll] = oacc[nt][r] / lrow[r];
      }
    }
  }
}

// ---------------------------------------------------------------------------
extern "C" void kernel_launch(void* const* d_in, const int* in_sizes, int n_in,
                              void* d_out, int out_size, void* d_ws, size_t ws_size,
                              hipStream_t stream) {
  const float* x  = (const float*)d_in[0];
  const float* wq = (const float*)d_in[1];
  const float* wk = (const float*)d_in[2];
  const float* wv = (const float*)d_in[3];

  const size_t nQ = (size_t)B_ * T_ * H_;        // 1M floats
  float* Q  = (float*)d_ws;
  float* KV = Q + nQ;                            // 2M floats, [B,T,128] interleaved

  proj_rope_kernel<<<(B_ * TT) / 4, 128, 0, stream>>>(x, wq, wk, wv, Q, KV);
  attn_kernel<<<TT, 64, 0, stream>>>(Q, KV, (float*)d_out);
}